// GCN_no_edge_attr_82300163326464
// MI455X (gfx1250) — compile-verified
//
#include <hip/hip_runtime.h>

typedef __attribute__((ext_vector_type(2))) float v2f;
typedef __attribute__((ext_vector_type(8))) float v8f;

// ---------------------------------------------------------------------------
// GCNConv: out = scatter_add( (xW)[src] * dinv[src]*dinv[dst] ) + (xW)[i]*dinv[i]^2 + b
// deg computed over self-loop-augmented dst (always >= 1).
// Strategy: CSR-by-dst to avoid 70M f32 atomics; y = (xW)*dinv[row] precomputed
// so aggregation is out[i] += dinv[i] * sum(y[src]).
// ---------------------------------------------------------------------------

__global__ void deg_init_kernel(int* __restrict__ deg, int n) {
  int i = blockIdx.x * blockDim.x + threadIdx.x;
  if (i < n) deg[i] = 1;  // self-loop
}

__global__ void deg_count_kernel(const int* __restrict__ dst, int* __restrict__ deg, int e) {
  int i = blockIdx.x * blockDim.x + threadIdx.x;
  if (i < e) atomicAdd(&deg[dst[i]], 1);
}

// Single-workgroup exclusive scan of (deg[i]-1) over N elements (chunked
// Hillis-Steele with running carry). Also emits dinv = rsqrt(deg) and a second
// copy of the offsets to be used as scatter cursors. rowstart has N+1 entries.
__global__ void scan_kernel(const int* __restrict__ deg, int* __restrict__ rowstart,
                            int* __restrict__ cursor, float* __restrict__ dinvf, int n) {
  __shared__ int buf[1024];
  __shared__ int carry_s;
  const int tid = threadIdx.x;
  if (tid == 0) carry_s = 0;
  __syncthreads();
  for (int base = 0; base < n; base += 1024) {
    int i = base + tid;
    int v = 0;
    if (i < n) {
      v = deg[i] - 1;                       // real in-edges (exclude self-loop)
      dinvf[i] = rsqrtf((float)(v + 1));    // dinv uses self-loop-augmented degree
    }
    buf[tid] = v;
    __syncthreads();
    #pragma unroll
    for (int off = 1; off < 1024; off <<= 1) {
      int t = (tid >= off) ? buf[tid - off] : 0;
      __syncthreads();
      buf[tid] += t;
      __syncthreads();
    }
    int incl = buf[tid];
    int c = carry_s;          // read old carry
    __syncthreads();          // everyone has read carry before it is updated
    if (i < n) {
      int o = c + incl - v;   // exclusive prefix
      rowstart[i] = o;
      cursor[i] = o;
    }
    if (tid == 0) carry_s = c + buf[1023];
    __syncthreads();
  }
  if (tid == 0) rowstart[n] = carry_s;      // == E
}

__global__ void scatter_kernel(const int* __restrict__ src, const int* __restrict__ dst,
                               int* __restrict__ cursor, int* __restrict__ csr, int e) {
  int i = blockIdx.x * blockDim.x + threadIdx.x;
  if (i < e) {
    int d = dst[i];
    int pos = atomicAdd(&cursor[d], 1);
    csr[pos] = src[i];
  }
}

// One wave per 16x16 output tile; K=64 via 16 chained V_WMMA_F32_16X16X4_F32.
// A layout (16x4 f32, 2 VGPR): lane t holds A[t&15][2*(t>>4) + {0,1}] -> contiguous float2.
// B layout (4x16 f32, 2 VGPR): lane t holds B[2*(t>>4) + {0,1}][t&15].
// C/D layout (16x16 f32, 8 VGPR): c[v] is element [8*(t>>4)+v][t&15].
// EXEC is all 1's (no divergence; grid exactly covers N/16 tiles).
__global__ void __launch_bounds__(128)
gemm_self_kernel(const float* __restrict__ x, const float* __restrict__ W,
                 const float* __restrict__ bias, const float* __restrict__ dinvf,
                 float* __restrict__ y, float* __restrict__ out) {
  const int t = threadIdx.x;            // lane 0..31
  const int m0 = blockIdx.x * 16;       // row tile
  const int n0 = threadIdx.y * 16;      // col tile (4 waves cover DOUT=64)
  const int l16 = t & 15;
  const int kh = (t >> 4) * 2;          // 0 or 2
  const int row = m0 + l16;
  const int col = n0 + l16;

  v8f c = {};
  #pragma unroll
  for (int k = 0; k < 64; k += 4) {
    v2f a = *(const v2f*)(x + row * 64 + k + kh);
    v2f b;
    b[0] = W[(k + kh) * 64 + col];
    b[1] = W[(k + kh + 1) * 64 + col];
    c = __builtin_amdgcn_wmma_f32_16x16x4_f32(
        /*neg_a=*/false, a, /*neg_b=*/false, b,
        /*c_mod=*/(short)0, c, /*reuse_a=*/false, /*reuse_b=*/false);
  }

  const float bv = bias[col];
  #pragma unroll
  for (int v = 0; v < 8; v++) {
    int m = m0 + 8 * (t >> 4) + v;
    float di = dinvf[m];
    float yv = c[v] * di;               // y = xW * dinv[row]
    y[m * 64 + col] = yv;
    out[m * 64 + col] = yv * di + bv;   // self-loop term + bias
  }
}

// Scalar fallback for row tails if N % 16 != 0 (not hit for N=100000).
__global__ void gemm_tail_kernel(const float* __restrict__ x, const float* __restrict__ W,
                                 const float* __restrict__ bias, const float* __restrict__ dinvf,
                                 float* __restrict__ y, float* __restrict__ out,
                                 int row0, int n) {
  int idx = blockIdx.x * blockDim.x + threadIdx.x;
  int row = row0 + idx / 64;
  int col = idx % 64;
  if (row >= n) return;
  float acc = 0.f;
  for (int k = 0; k < 64; k++) acc += x[row * 64 + k] * W[k * 64 + col];
  float di = dinvf[row];
  float yv = acc * di;
  y[row * 64 + col] = yv;
  out[row * 64 + col] = yv * di + bias[col];
}

// One wave per node: lane t owns cols {2t, 2t+1}; walk CSR in-edges, sum y[src]
// rows (float2 loads, mostly L2 hits since y is 25.6MB < 192MB L2), then one
// non-atomic read-modify-write of out.
__global__ void __launch_bounds__(256)
aggregate_kernel(const int* __restrict__ rowstart, const int* __restrict__ csr,
                 const float* __restrict__ y, const float* __restrict__ dinvf,
                 float* __restrict__ out, int n) {
  int node = blockIdx.x * blockDim.y + threadIdx.y;
  if (node >= n) return;
  int t = threadIdx.x;
  int beg = rowstart[node];
  int end = rowstart[node + 1];
  float ax = 0.f, ay = 0.f;
  for (int j = beg; j < end; j++) {
    int s = csr[j];
    v2f v = *(const v2f*)(y + s * 64 + 2 * t);
    ax += v[0];
    ay += v[1];
  }
  float di = dinvf[node];
  int o = node * 64 + 2 * t;
  out[o]     += di * ax;
  out[o + 1] += di * ay;
}

static inline size_t ws_align(size_t x) { return (x + 255) & ~(size_t)255; }

extern "C" void kernel_launch(void* const* d_in, const int* in_sizes, int n_in,
                              void* d_out, int out_size, void* d_ws, size_t ws_size,
                              hipStream_t stream) {
  const float* x  = (const float*)d_in[0];
  const int*   ei = (const int*)d_in[1];
  const float* W  = (const float*)d_in[2];
  const float* b  = (const float*)d_in[3];
  float* out = (float*)d_out;

  const int N = in_sizes[0] / 64;
  const int E = in_sizes[1] / 2;
  const int* src = ei;       // edge_index[0]
  const int* dst = ei + E;   // edge_index[1]

  // Workspace layout (~31.2 MB): deg | dinv | rowstart | cursor | csr | y
  char* ws = (char*)d_ws;
  size_t off = 0;
  int*   deg      = (int*)(ws + off);   off = ws_align(off + (size_t)N * sizeof(int));
  float* dinvf    = (float*)(ws + off); off = ws_align(off + (size_t)N * sizeof(float));
  int*   rowstart = (int*)(ws + off);   off = ws_align(off + (size_t)(N + 1) * sizeof(int));
  int*   cursor   = (int*)(ws + off);   off = ws_align(off + (size_t)N * sizeof(int));
  int*   csr      = (int*)(ws + off);   off = ws_align(off + (size_t)E * sizeof(int));
  float* y        = (float*)(ws + off); off = ws_align(off + (size_t)N * 64 * sizeof(float));

  deg_init_kernel<<<(N + 255) / 256, 256, 0, stream>>>(deg, N);
  deg_count_kernel<<<(E + 255) / 256, 256, 0, stream>>>(dst, deg, E);
  scan_kernel<<<1, 1024, 0, stream>>>(deg, rowstart, cursor, dinvf, N);
  scatter_kernel<<<(E + 255) / 256, 256, 0, stream>>>(src, dst, cursor, csr, E);

  int mtiles = N / 16;
  if (mtiles > 0)
    gemm_self_kernel<<<mtiles, dim3(32, 4), 0, stream>>>(x, W, b, dinvf, y, out);
  int rem = N - mtiles * 16;
  if (rem > 0)
    gemm_tail_kernel<<<(rem * 64 + 255) / 256, 256, 0, stream>>>(x, W, b, dinvf, y, out,
                                                                 mtiles * 16, N);

  aggregate_kernel<<<(N + 7) / 8, dim3(32, 8), 0, stream>>>(rowstart, csr, y, dinvf, out, N);
}